// MambaEncoder_23141283791521
// MI455X (gfx1250) — compile-verified
//
#include <hip/hip_runtime.h>
#include <hip/hip_bf16.h>

// ---------------------------------------------------------------------------
// MI455X / gfx1250 Mamba encoder.
// - GEMMs: V_WMMA_F32_16X16X4_F32, 128x64 workgroup tile, 32x32 per wave
//   (4 accumulators), LDS-staged K-chunks, global_prefetch of next chunk.
// - Scan: serial recurrence with latency amortization -- 64-step B/C LDS
//   staging + 8-step register buffering of dt/u/z.
// ---------------------------------------------------------------------------

typedef __attribute__((ext_vector_type(2))) float v2f;
typedef __attribute__((ext_vector_type(8))) float v8f;

#define KB 16          // K-chunk staged in LDS
#define LDS_PAD 20     // row stride (floats): 16B-aligned float4 stores, conflict-free frag reads

__device__ __forceinline__ float siluf(float x) {
    return x / (1.f + __expf(-x));
}
__device__ __forceinline__ float softplusf(float x) {
    return (x > 20.f) ? x : log1pf(__expf(x));
}
__device__ __forceinline__ float apply_act(float x, int act, float b) {
    if (act == 1) return siluf(x);
    if (act == 2) return softplusf(x + b);
    return x;
}

// O[M,N] = act( A[M,K(lda)] * W[N,K]^T (+bias) )
// BM=128, BN=64 per workgroup; 8 waves, each wave owns a 32x32 slab
// (four 16x16 f32 WMMA accumulators: 4 MACs per 4 fragment loads).
__global__ __launch_bounds__(256)
void gemm_wmma_f32(const float* __restrict__ A, int lda,
                   const float* __restrict__ W,
                   const float* __restrict__ bias,
                   float* __restrict__ O, int ldo,
                   int M, int N, int K, int act)
{
    __shared__ float As[128][LDS_PAD];
    __shared__ float Ws[64][LDS_PAD];

    const int tid  = threadIdx.x;
    const int wave = tid >> 5;
    const int lane = tid & 31;
    const int m0 = blockIdx.y * 128;
    const int n0 = blockIdx.x * 64;
    const int mw = (wave & 3) * 32;   // wave's M offset within tile
    const int nw = (wave >> 2) * 32;  // wave's N offset within tile

    v8f c00 = {}, c01 = {}, c10 = {}, c11 = {};

    // staging indices: W tile 64x16 = 1 float4/thread; A tile 128x16 = 2/thread
    const int wr = tid >> 2;          // 0..63
    const int wc = (tid & 3) * 4;     // 0,4,8,12

    for (int k = 0; k < K; k += KB) {
        __syncthreads();
        // Stage A tile (128 x 16): two float4 per thread, coalesced
#pragma unroll
        for (int s = 0; s < 2; s++) {
            const int idx = tid + s * 256;     // 0..511
            const int r = idx >> 2;            // 0..127
            const int c = (idx & 3) * 4;
            float4 v = make_float4(0.f, 0.f, 0.f, 0.f);
            if (m0 + r < M)
                v = *reinterpret_cast<const float4*>(A + (size_t)(m0 + r) * lda + k + c);
            *reinterpret_cast<float4*>(&As[r][c]) = v;
        }
        // Stage W tile (64 x 16)
        {
            float4 v = make_float4(0.f, 0.f, 0.f, 0.f);
            if (n0 + wr < N)
                v = *reinterpret_cast<const float4*>(W + (size_t)(n0 + wr) * K + k + wc);
            *reinterpret_cast<float4*>(&Ws[wr][wc]) = v;
        }
        // Prefetch next K-chunk into cache while this chunk computes
        if (k + KB < K) {
            if (m0 + wr < M)
                __builtin_prefetch(A + (size_t)(m0 + wr) * lda + k + KB + wc, 0, 3);
            if (n0 + wr < N)
                __builtin_prefetch(W + (size_t)(n0 + wr) * K + k + KB + wc, 0, 3);
        }
        __syncthreads();

        const int koff = (lane < 16) ? 0 : 2;  // ISA 16x4 f32 A layout: half-wave K split
        const int mr0  = mw + (lane & 15);
        const int mr1  = mr0 + 16;
        const int nc0  = nw + (lane & 15);
        const int nc1  = nc0 + 16;

#pragma unroll
        for (int kk = 0; kk < KB; kk += 4) {
            v2f a0; a0.x = As[mr0][kk + koff]; a0.y = As[mr0][kk + koff + 1];
            v2f a1; a1.x = As[mr1][kk + koff]; a1.y = As[mr1][kk + koff + 1];
            v2f b0; b0.x = Ws[nc0][kk + koff]; b0.y = Ws[nc0][kk + koff + 1];
            v2f b1; b1.x = Ws[nc1][kk + koff]; b1.y = Ws[nc1][kk + koff + 1];
            c00 = __builtin_amdgcn_wmma_f32_16x16x4_f32(false, a0, false, b0,
                                                        (short)0, c00, false, false);
            c01 = __builtin_amdgcn_wmma_f32_16x16x4_f32(false, a0, false, b1,
                                                        (short)0, c01, false, false);
            c10 = __builtin_amdgcn_wmma_f32_16x16x4_f32(false, a1, false, b0,
                                                        (short)0, c10, false, false);
            c11 = __builtin_amdgcn_wmma_f32_16x16x4_f32(false, a1, false, b1,
                                                        (short)0, c11, false, false);
        }
    }

    // Epilogue: C/D layout -- lane l, VGPR v : M = v + 8*(l>=16), N = l%16
    const int col0 = n0 + nw + (lane & 15);
    const int col1 = col0 + 16;
    const float b0v = (act == 2 && col0 < N) ? bias[col0] : 0.f;
    const float b1v = (act == 2 && col1 < N) ? bias[col1] : 0.f;
    const int r0 = m0 + mw + ((lane >> 4) << 3);

#pragma unroll
    for (int v = 0; v < 8; v++) {
        const int row0 = r0 + v;
        const int row1 = row0 + 16;
        if (row0 < M) {
            if (col0 < N) O[(size_t)row0 * ldo + col0] = apply_act(c00[v], act, b0v);
            if (col1 < N) O[(size_t)row0 * ldo + col1] = apply_act(c01[v], act, b1v);
        }
        if (row1 < M) {
            if (col0 < N) O[(size_t)row1 * ldo + col0] = apply_act(c10[v], act, b0v);
            if (col1 < N) O[(size_t)row1 * ldo + col1] = apply_act(c11[v], act, b1v);
        }
    }
}

// h[b,l,d] = x[b,l] * emb_w[d] + emb_b[d]
__global__ __launch_bounds__(256)
void embed_kernel(const float* __restrict__ x, const float* __restrict__ ew,
                  const float* __restrict__ eb, float* __restrict__ h)
{
    const int idx = blockIdx.x * 256 + threadIdx.x;   // over (B*L) * 1024
    const int d   = idx & 1023;
    const int bl  = idx >> 10;
    h[idx] = x[bl] * ew[d] + eb[d];
}

// u = silu( conv_b + depthwise causal 4-tap conv over xx = xz[..., :2048] )
__global__ __launch_bounds__(256)
void conv_silu_kernel(const float* __restrict__ xz, const float* __restrict__ cw,
                      const float* __restrict__ cb, float* __restrict__ u)
{
    const int L = 1024, DI = 2048;
    const int idx = blockIdx.x * 256 + threadIdx.x;   // over B*L*DI
    const int d   = idx & (DI - 1);
    const int bl  = idx >> 11;
    const int l   = bl & (L - 1);
    const int b   = bl >> 10;

    float acc = cb[d];
#pragma unroll
    for (int k = 0; k < 4; k++) {
        const int ls = l - 3 + k;
        float xv = 0.f;
        if (ls >= 0)
            xv = xz[((size_t)(b * L + ls)) * 4096 + d];  // xx half of xz
        acc += cw[d * 4 + k] * xv;
    }
    u[idx] = siluf(acc);
}

// Selective scan: one thread per (b, d) channel; 16 states in VGPRs.
// Latency amortization: B_t/C_t for 64 timesteps staged in LDS per barrier
// pair; dt/u/z buffered 8 timesteps deep in registers so loads overlap.
// Fuses y = (scan_y + D*u) * silu(z).
#define SCAN_TB 64
#define SCAN_TR 8
__global__ __launch_bounds__(256)
void scan_kernel(const float* __restrict__ xdbl, const float* __restrict__ dt,
                 const float* __restrict__ u, const float* __restrict__ xz,
                 const float* __restrict__ A_log, const float* __restrict__ Dp,
                 float* __restrict__ y)
{
    const int L = 1024, DI = 2048;
    const int tid = threadIdx.x;
    const int b   = blockIdx.x >> 3;
    const int d   = ((blockIdx.x & 7) << 8) + tid;

    __shared__ float sB[SCAN_TB][16];
    __shared__ float sC[SCAN_TB][16];

    float Arow[16];
    float st[16];
#pragma unroll
    for (int n = 0; n < 16; n++) {
        Arow[n] = -__expf(A_log[d * 16 + n]);
        st[n]   = 0.f;
    }
    const float Dd = Dp[d];
    const size_t base = (size_t)b * L;

    for (int tc = 0; tc < L; tc += SCAN_TB) {
        __syncthreads();
        // Cooperative stage of B/C for 64 timesteps: 2048 floats / 256 threads
#pragma unroll
        for (int s = 0; s < (SCAN_TB * 32) / 256; s++) {
            const int j  = tid + s * 256;
            const int tt = j >> 5;
            const int e  = j & 31;
            const float v = xdbl[(base + tc + tt) * 96 + 64 + e];
            if (e < 16) sB[tt][e] = v;
            else        sC[tt][e - 16] = v;
        }
        __syncthreads();

        for (int tr = 0; tr < SCAN_TB; tr += SCAN_TR) {
            float dtv[SCAN_TR], uv[SCAN_TR], zv[SCAN_TR], yr[SCAN_TR];
            // Issue all loads for this 8-step window up front (overlap latency)
#pragma unroll
            for (int j = 0; j < SCAN_TR; j++) {
                const size_t row = base + tc + tr + j;
                dtv[j] = dt[row * DI + d];
                uv[j]  = u[row * DI + d];
                zv[j]  = xz[row * 4096 + 2048 + d];
            }
#pragma unroll
            for (int j = 0; j < SCAN_TR; j++) {
                const int tt = tr + j;
                const float du = dtv[j] * uv[j];
                float acc = 0.f;
#pragma unroll
                for (int n = 0; n < 16; n++) {
                    const float dA = __expf(dtv[j] * Arow[n]);
                    st[n] = dA * st[n] + du * sB[tt][n];
                    acc  += st[n] * sC[tt][n];
                }
                yr[j] = (acc + Dd * uv[j]) * siluf(zv[j]);
            }
#pragma unroll
            for (int j = 0; j < SCAN_TR; j++)
                y[(base + tc + tr + j) * DI + d] = yr[j];
        }
    }
}

extern "C" void kernel_launch(void* const* d_in, const int* in_sizes, int n_in,
                              void* d_out, int out_size, void* d_ws, size_t ws_size,
                              hipStream_t stream) {
    (void)in_sizes; (void)n_in; (void)out_size; (void)ws_size;

    const float* x     = (const float*)d_in[0];
    const float* emb_w = (const float*)d_in[1];
    const float* emb_b = (const float*)d_in[2];
    const float* inw   = (const float*)d_in[3];
    const float* cw    = (const float*)d_in[4];
    const float* cb    = (const float*)d_in[5];
    const float* xpw   = (const float*)d_in[6];
    const float* dtw   = (const float*)d_in[7];
    const float* dtb   = (const float*)d_in[8];
    const float* Alog  = (const float*)d_in[9];
    const float* Dpar  = (const float*)d_in[10];
    const float* outw  = (const float*)d_in[11];

    const int Bc = 2, L = 1024, DM = 1024, DI = 2048, M = Bc * L;  // M = 2048

    float* ws   = (float*)d_ws;
    float* h    = ws;                    // 2,097,152
    float* xz   = h    + 2097152;        // 8,388,608
    float* u    = xz   + 8388608;        // 4,194,304
    float* xdbl = u    + 4194304;        //   196,608
    float* dt   = xdbl + 196608;         // 4,194,304
    float* y    = dt   + 4194304;        // 4,194,304

    embed_kernel<<<(Bc * L * DM) / 256, 256, 0, stream>>>(x, emb_w, emb_b, h);

    for (int i = 0; i < 4; i++) {
        const float* inw_i  = inw  + (size_t)i * 4096 * 1024;
        const float* cw_i   = cw   + (size_t)i * DI * 4;
        const float* cb_i   = cb   + (size_t)i * DI;
        const float* xpw_i  = xpw  + (size_t)i * 96 * DI;
        const float* dtw_i  = dtw  + (size_t)i * DI * 64;
        const float* dtb_i  = dtb  + (size_t)i * DI;
        const float* Alog_i = Alog + (size_t)i * DI * 16;
        const float* Dp_i   = Dpar + (size_t)i * DI;
        const float* outw_i = outw + (size_t)i * 1024 * DI;

        // xz = h @ in_proj^T              (2048 x 1024) x (4096 x 1024)^T
        {
            dim3 g(4096 / 64, M / 128);
            gemm_wmma_f32<<<g, 256, 0, stream>>>(h, DM, inw_i, nullptr,
                                                 xz, 4096, M, 4096, DM, 0);
        }
        // u = silu(causal depthwise conv(xx) + cb)
        conv_silu_kernel<<<(Bc * L * DI) / 256, 256, 0, stream>>>(xz, cw_i, cb_i, u);
        // x_dbl = u @ x_proj^T            (2048 x 2048) x (96 x 2048)^T
        {
            dim3 g((96 + 63) / 64, M / 128);
            gemm_wmma_f32<<<g, 256, 0, stream>>>(u, DI, xpw_i, nullptr,
                                                 xdbl, 96, M, 96, DI, 0);
        }
        // dt = softplus(dt_lo @ dt_proj^T + dt_b)   (2048 x 64) x (2048 x 64)^T
        {
            dim3 g(DI / 64, M / 128);
            gemm_wmma_f32<<<g, 256, 0, stream>>>(xdbl, 96, dtw_i, dtb_i,
                                                 dt, DI, M, DI, 64, 2);
        }
        // selective scan + D*u + silu(z) gate
        scan_kernel<<<16, 256, 0, stream>>>(xdbl, dt, u, xz, Alog_i, Dp_i, y);
        // h' = silu(y @ out_proj^T)       (2048 x 2048) x (1024 x 2048)^T
        {
            float* dst = (i == 3) ? (float*)d_out : h;
            dim3 g(1024 / 64, M / 128);
            gemm_wmma_f32<<<g, 256, 0, stream>>>(y, DI, outw_i, nullptr,
                                                 dst, 1024, M, 1024, DI, 1);
        }
    }
}